// MLP_model_90598040142266
// MI455X (gfx1250) — compile-verified
//
#include <hip/hip_runtime.h>
#include <hip/hip_bf16.h>

// ---------------------------------------------------------------------------
// MLP recommender forward on MI455X (gfx1250), fp32 via V_WMMA_F32_16X16X4_F32
//
//  movie_emb = movie_table @ Wm + bm            [100000,256]  (workspace, L2-resident)
//  u0        = users @ Wu + bu                  [16384,256]   (workspace)
//  u0       += mean(movie_emb[nbr_ids], axis=1)               (gather, L2 hits)
//  x rows    = { u0 | movie_emb[pos] | movie_emb[neg] }  -> 49152 rows
//  out       = relu(relu(x@W1+b1)@W2+b2)        [49152,128] = d_out
//
// Tiling: 256 threads = 8 wave32 (2M x 4N).  BM=64, BK=32.
// Wave computes 32M x 64N (or 32N in layer 2) with f32 16x16x4 WMMA.
// W staged K-pair-interleaved in LDS: Wp[k/2][2n + (k&1)], stride 544 floats,
// so every B fragment is ONE even-aligned ds_load_b64, conflict-free.
// Global->LDS staging is double-buffered through registers.
// ---------------------------------------------------------------------------

typedef __attribute__((ext_vector_type(2))) float v2f;
typedef __attribute__((ext_vector_type(8))) float v8f;

#define B_ 16384
#define NMOV 100000
#define NNBR 50

#define WPSTRIDE 544   // floats per Wp row: 512 data + 32 pad (half-wave -> +32 banks)

__device__ __forceinline__ v8f wmma4(v2f a, v2f b, v8f c) {
  // 8 args: (neg_a, A, neg_b, B, c_mod, C, reuse_a, reuse_b)
  return __builtin_amdgcn_wmma_f32_16x16x4_f32(false, a, false, b, (short)0, c,
                                               false, false);
}

__device__ __forceinline__ v8f vzero8() {
  return (v8f){0.f, 0.f, 0.f, 0.f, 0.f, 0.f, 0.f, 0.f};
}

// ---------------------------------------------------------------------------
// C[M,256] = A[M,K] @ W[K,256] + bias.  grid.x = ceil(M/64). Row-guarded.
// ---------------------------------------------------------------------------
__global__ __launch_bounds__(256) void gemm_n256(const float* __restrict__ A,
                                                 const float* __restrict__ W,
                                                 const float* __restrict__ bias,
                                                 float* __restrict__ C,
                                                 int M, int K) {
  __shared__ float As[64][36];           // 9.2 KB, b64 reads conflict-spread
  __shared__ float Wp[16][WPSTRIDE];     // 34.8 KB, K-pair interleaved

  const int tid  = threadIdx.x;
  const int lane = tid & 31;
  const int warp = tid >> 5;
  const int wm   = warp >> 2;            // 0..1 -> 32 rows each
  const int wn   = warp & 3;             // 0..3 -> 64 cols each
  const int mb   = wm << 5;
  const int hl   = lane >> 4;
  const int ml   = lane & 15;

  const int rowBase = blockIdx.x * 64;

  v8f acc[2][4];
#pragma unroll
  for (int m = 0; m < 2; ++m)
#pragma unroll
    for (int j = 0; j < 4; ++j) acc[m][j] = vzero8();

  // A staging: 64x32 floats, 2 float4/thread.  r = f4>>3, c4 = (f4&7)*4
  // W staging: 32x256 floats as 2x2 units: u -> k2=u>>7, n2=u&127 (8 units/thread)
  float4 aPre[2];
  float2 wPreLo[8], wPreHi[8];

  auto loadA = [&](int kk0) {
#pragma unroll
    for (int i = 0; i < 2; ++i) {
      const int f4 = tid + (i << 8);
      const int r  = f4 >> 3;
      const int c4 = (f4 & 7) << 2;
      const int row = rowBase + r;
      if (row < M)
        aPre[i] = *(const float4*)(A + (size_t)row * K + kk0 + c4);
      else
        aPre[i] = make_float4(0.f, 0.f, 0.f, 0.f);
    }
  };
  auto loadW = [&](int kk0) {
#pragma unroll
    for (int i = 0; i < 8; ++i) {
      const int u  = tid + (i << 8);
      const int k2 = u >> 7;
      const int n2 = u & 127;
      const float* p = W + (size_t)(kk0 + (k2 << 1)) * 256 + (n2 << 1);
      wPreLo[i] = *(const float2*)p;
      wPreHi[i] = *(const float2*)(p + 256);
    }
  };
  auto stageAW = [&]() {
#pragma unroll
    for (int i = 0; i < 2; ++i) {
      const int f4 = tid + (i << 8);
      *(float4*)&As[f4 >> 3][(f4 & 7) << 2] = aPre[i];
    }
#pragma unroll
    for (int i = 0; i < 8; ++i) {
      const int u  = tid + (i << 8);
      const int k2 = u >> 7;
      const int n2 = u & 127;
      // interleave: [k,n]->2n, [k+1,n]->2n+1, [k,n+1]->2n+2, [k+1,n+1]->2n+3
      *(float4*)&Wp[k2][n2 << 2] =
          make_float4(wPreLo[i].x, wPreHi[i].x, wPreLo[i].y, wPreHi[i].y);
    }
  };

  loadA(0);
  loadW(0);
#pragma unroll 1
  for (int kk0 = 0; kk0 < K; kk0 += 32) {
    stageAW();
    __syncthreads();
    if (kk0 + 32 < K) {       // prefetch next chunk while WMMAs run
      loadA(kk0 + 32);
      loadW(kk0 + 32);
    }
#pragma unroll
    for (int kk = 0; kk < 32; kk += 4) {
      const int k0 = kk + (hl << 1);
      const int kp = (kk >> 1) + hl;
      const v2f a0 = *(const v2f*)&As[mb + ml][k0];
      const v2f a1 = *(const v2f*)&As[mb + 16 + ml][k0];
#pragma unroll
      for (int j = 0; j < 4; ++j) {
        const int n = (wn << 6) + (j << 4) + ml;
        const v2f b = *(const v2f*)&Wp[kp][n << 1];
        acc[0][j] = wmma4(a0, b, acc[0][j]);
        acc[1][j] = wmma4(a1, b, acc[1][j]);
      }
    }
    __syncthreads();
  }

  // epilogue: acc[m][j] VGPR i -> (M = mb + 16m + 8*hl + i, N = 64*wn + 16j + ml)
#pragma unroll
  for (int m = 0; m < 2; ++m)
#pragma unroll
    for (int j = 0; j < 4; ++j) {
      const int col = (wn << 6) + (j << 4) + ml;
      const float bv = bias[col];
#pragma unroll
      for (int i = 0; i < 8; ++i) {
        const int row = rowBase + mb + (m << 4) + (hl << 3) + i;
        if (row < M) C[(size_t)row * 256 + col] = acc[m][j][i] + bv;
      }
    }
}

// ---------------------------------------------------------------------------
// xu[u,:] += mean over 50 neighbors of movie_emb[nbr_ids[u,t], :]   (in place)
// 4 users/block, 64 threads/user, float4/thread.  grid = 16384/4.
// ---------------------------------------------------------------------------
__global__ __launch_bounds__(256) void nbr_mean_add(float* __restrict__ xu,
                                                    const float* __restrict__ memb,
                                                    const int* __restrict__ nbr) {
  const int u  = blockIdx.x * 4 + (threadIdx.x >> 6);
  const int c4 = (threadIdx.x & 63) << 2;
  const int* ids = nbr + (size_t)u * NNBR;

  float4 acc = make_float4(0.f, 0.f, 0.f, 0.f);
#pragma unroll 5
  for (int t = 0; t < NNBR; ++t) {
    if (t + 4 < NNBR)  // hide L2 gather latency (global_prefetch_b8)
      __builtin_prefetch(memb + (size_t)ids[t + 4] * 256 + c4, 0, 0);
    const float4 v = *(const float4*)(memb + (size_t)ids[t] * 256 + c4);
    acc.x += v.x; acc.y += v.y; acc.z += v.z; acc.w += v.w;
  }
  const float s = 1.0f / (float)NNBR;
  float4* p = (float4*)(xu + (size_t)u * 256 + c4);
  float4 o = *p;
  o.x += acc.x * s; o.y += acc.y * s; o.z += acc.z * s; o.w += acc.w * s;
  *p = o;
}

// ---------------------------------------------------------------------------
// Fused MLP over 3*16384 stacked rows, 64 rows/block (grid = 768):
//   phase 1: gather x tile -> LDS, h1 = relu(x@W1+b1)  (K=256,N=256), h1 in LDS
//   phase 2: out = relu(h1@W2+b2)                      (K=256,N=128) -> d_out
// ---------------------------------------------------------------------------
__global__ __launch_bounds__(256) void fused_mlp(const float* __restrict__ xu,
                                                 const float* __restrict__ memb,
                                                 const int* __restrict__ pos_ids,
                                                 const int* __restrict__ neg_ids,
                                                 const float* __restrict__ W1,
                                                 const float* __restrict__ b1,
                                                 const float* __restrict__ W2,
                                                 const float* __restrict__ b2,
                                                 float* __restrict__ out) {
  __shared__ float Xs[64][36];          //  9.2 KB
  __shared__ float Wp[16][WPSTRIDE];    // 34.8 KB
  __shared__ float H1[64][276];         // 70.7 KB, b64-read conflict-free stride

  const int tid  = threadIdx.x;
  const int lane = tid & 31;
  const int warp = tid >> 5;
  const int wm   = warp >> 2;
  const int wn   = warp & 3;
  const int mb   = wm << 5;
  const int hl   = lane >> 4;
  const int ml   = lane & 15;

  const int rowBase = blockIdx.x * 64;

  float4 aPre[2];
  float2 wPreLo[8], wPreHi[8];

  auto loadX = [&](int kk0) {
#pragma unroll
    for (int i = 0; i < 2; ++i) {
      const int f4 = tid + (i << 8);
      const int r  = f4 >> 3;
      const int c4 = (f4 & 7) << 2;
      const int gr  = rowBase + r;         // 0..49151
      const int seg = gr >> 14;
      const int rr  = gr & (B_ - 1);
      const float* src;
      if (seg == 0)      src = xu   + (size_t)rr * 256;
      else if (seg == 1) src = memb + (size_t)pos_ids[rr] * 256;
      else               src = memb + (size_t)neg_ids[rr] * 256;
      aPre[i] = *(const float4*)(src + kk0 + c4);
    }
  };
  auto loadW1 = [&](int kk0) {
#pragma unroll
    for (int i = 0; i < 8; ++i) {
      const int u  = tid + (i << 8);
      const int k2 = u >> 7;
      const int n2 = u & 127;
      const float* p = W1 + (size_t)(kk0 + (k2 << 1)) * 256 + (n2 << 1);
      wPreLo[i] = *(const float2*)p;
      wPreHi[i] = *(const float2*)(p + 256);
    }
  };
  auto stageXW1 = [&]() {
#pragma unroll
    for (int i = 0; i < 2; ++i) {
      const int f4 = tid + (i << 8);
      *(float4*)&Xs[f4 >> 3][(f4 & 7) << 2] = aPre[i];
    }
#pragma unroll
    for (int i = 0; i < 8; ++i) {
      const int u  = tid + (i << 8);
      *(float4*)&Wp[u >> 7][(u & 127) << 2] =
          make_float4(wPreLo[i].x, wPreHi[i].x, wPreLo[i].y, wPreHi[i].y);
    }
  };

  // ---- phase 1: h1 = relu(x @ W1 + b1), K=256, N=256 ----
  v8f acc[2][4];
#pragma unroll
  for (int m = 0; m < 2; ++m)
#pragma unroll
    for (int j = 0; j < 4; ++j) acc[m][j] = vzero8();

  loadX(0);
  loadW1(0);
#pragma unroll 1
  for (int kk0 = 0; kk0 < 256; kk0 += 32) {
    stageXW1();
    __syncthreads();
    if (kk0 + 32 < 256) {
      loadX(kk0 + 32);
      loadW1(kk0 + 32);
    }
#pragma unroll
    for (int kk = 0; kk < 32; kk += 4) {
      const int k0 = kk + (hl << 1);
      const int kp = (kk >> 1) + hl;
      const v2f a0 = *(const v2f*)&Xs[mb + ml][k0];
      const v2f a1 = *(const v2f*)&Xs[mb + 16 + ml][k0];
#pragma unroll
      for (int j = 0; j < 4; ++j) {
        const int n = (wn << 6) + (j << 4) + ml;
        const v2f b = *(const v2f*)&Wp[kp][n << 1];
        acc[0][j] = wmma4(a0, b, acc[0][j]);
        acc[1][j] = wmma4(a1, b, acc[1][j]);
      }
    }
    __syncthreads();
  }

  // bias + relu -> H1 (LDS)
#pragma unroll
  for (int m = 0; m < 2; ++m)
#pragma unroll
    for (int j = 0; j < 4; ++j) {
      const int col = (wn << 6) + (j << 4) + ml;
      const float bv = b1[col];
#pragma unroll
      for (int i = 0; i < 8; ++i)
        H1[mb + (m << 4) + (hl << 3) + i][col] = fmaxf(acc[m][j][i] + bv, 0.f);
    }
  __syncthreads();

  // ---- phase 2: out = relu(h1 @ W2 + b2), K=256, N=128 ----
  // wave = 32M x 32N.  W2 staged in Wp: 32x128 as 2x2 units, 4 units/thread.
  v8f acc2[2][2];
#pragma unroll
  for (int m = 0; m < 2; ++m)
#pragma unroll
    for (int j = 0; j < 2; ++j) acc2[m][j] = vzero8();

  auto loadW2 = [&](int kk0) {
#pragma unroll
    for (int i = 0; i < 4; ++i) {
      const int u  = tid + (i << 8);
      const int k2 = u >> 6;
      const int n2 = u & 63;
      const float* p = W2 + (size_t)(kk0 + (k2 << 1)) * 128 + (n2 << 1);
      wPreLo[i] = *(const float2*)p;
      wPreHi[i] = *(const float2*)(p + 128);
    }
  };
  auto stageW2 = [&]() {
#pragma unroll
    for (int i = 0; i < 4; ++i) {
      const int u = tid + (i << 8);
      *(float4*)&Wp[u >> 6][(u & 63) << 2] =
          make_float4(wPreLo[i].x, wPreHi[i].x, wPreLo[i].y, wPreHi[i].y);
    }
  };

  loadW2(0);
#pragma unroll 1
  for (int kk0 = 0; kk0 < 256; kk0 += 32) {
    stageW2();
    __syncthreads();
    if (kk0 + 32 < 256) loadW2(kk0 + 32);
#pragma unroll
    for (int kk = 0; kk < 32; kk += 4) {
      const int k0 = kk + (hl << 1);
      const int kp = (kk >> 1) + hl;
      const v2f a0 = *(const v2f*)&H1[mb + ml][kk0 + k0];
      const v2f a1 = *(const v2f*)&H1[mb + 16 + ml][kk0 + k0];
#pragma unroll
      for (int j = 0; j < 2; ++j) {
        const int n = (wn << 5) + (j << 4) + ml;
        const v2f b = *(const v2f*)&Wp[kp][n << 1];
        acc2[0][j] = wmma4(a0, b, acc2[0][j]);
        acc2[1][j] = wmma4(a1, b, acc2[1][j]);
      }
    }
    __syncthreads();
  }

#pragma unroll
  for (int m = 0; m < 2; ++m)
#pragma unroll
    for (int j = 0; j < 2; ++j) {
      const int col = (wn << 5) + (j << 4) + ml;
      const float bv = b2[col];
#pragma unroll
      for (int i = 0; i < 8; ++i) {
        const size_t row = (size_t)rowBase + mb + (m << 4) + (hl << 3) + i;
        out[row * 128 + col] = fmaxf(acc2[m][j][i] + bv, 0.f);
      }
    }
}

// ---------------------------------------------------------------------------
extern "C" void kernel_launch(void* const* d_in, const int* in_sizes, int n_in,
                              void* d_out, int out_size, void* d_ws, size_t ws_size,
                              hipStream_t stream) {
  (void)in_sizes; (void)n_in; (void)out_size; (void)ws_size;

  const float* users       = (const float*)d_in[0];
  // d_in[1] pos_movies, d_in[2] neg_movies, d_in[3] user_ids: unused by reference
  const int*   pos_ids     = (const int*)d_in[4];
  const int*   neg_ids     = (const int*)d_in[5];
  const int*   nbr_ids     = (const int*)d_in[6];
  const float* movie_table = (const float*)d_in[7];
  const float* Wu          = (const float*)d_in[8];
  const float* bu          = (const float*)d_in[9];
  const float* Wm          = (const float*)d_in[10];
  const float* bm          = (const float*)d_in[11];
  const float* W1          = (const float*)d_in[12];
  const float* b1          = (const float*)d_in[13];
  const float* W2          = (const float*)d_in[14];
  const float* b2          = (const float*)d_in[15];

  float* memb = (float*)d_ws;                        // [100000,256] = 102.4 MB
  float* xu   = memb + (size_t)NMOV * 256;           // [16384,256]  =  16.8 MB

  dim3 blk(256);
  // movie_emb = movie_table @ Wm + bm   (M=100000 -> 1563 row-guarded blocks)
  gemm_n256<<<(NMOV + 63) / 64, blk, 0, stream>>>(movie_table, Wm, bm, memb,
                                                  NMOV, 128);
  // u0 = users @ Wu + bu
  gemm_n256<<<B_ / 64, blk, 0, stream>>>(users, Wu, bu, xu, B_, 128);
  // u0 += mean(movie_emb[nbr_ids])
  nbr_mean_add<<<B_ / 4, blk, 0, stream>>>(xu, memb, nbr_ids);
  // stacked MLP -> d_out
  fused_mlp<<<(3 * B_) / 64, blk, 0, stream>>>(xu, memb, pos_ids, neg_ids,
                                               W1, b1, W2, b2, (float*)d_out);
}